// WarpingLoss_51410758533564
// MI455X (gfx1250) — compile-verified
//
#include <hip/hip_runtime.h>

// WarpingLoss for MI455X (gfx1250, wave32).
// loss = mean|warp(L,flow_lm)-gt| + mean|warp(R,flow_rm)-gt|
// Memory-bound gather (~383 MB unique -> ~16.5 us at 23.3 TB/s).
// Reduction uses V_WMMA_F32_16X16X4_F32 with an all-ones B matrix as a
// deterministic f32 row-sum engine (D = A*1 + C).

typedef __attribute__((ext_vector_type(2))) float v2f;
typedef __attribute__((ext_vector_type(8))) float v8f;

#define BDIM 256

static constexpr int kB = 8, kC = 3, kH = 720, kW = 1280;
static constexpr int kHW = kH * kW;                       // 921600
static constexpr int kNPIX = kB * kHW;                    // 7372800
static constexpr long long kN = (long long)kB * kC * kHW; // 22118400
static constexpr int kNBLK = 7200;                        // 7200*256*4 == kNPIX

// ---------------------------------------------------------------------------
// Block-wide sum via WMMA f32 16x16x4 with B = ones.
// Each wmma consumes 64 fp32 partials (A is 16x4 striped over 32 lanes,
// 2 VGPRs/lane) and accumulates row sums into C. Since B is all-ones, the
// result is the exact sum of all 64 A values regardless of lane->(m,k)
// placement details. 256 values -> 4 accumulating WMMAs -> 16 row sums
// (replicated across columns) -> 8 in-lane adds + one cross-half shuffle.
// EXEC is all-ones inside wave 0 (tid<32 is exactly one wave32).
// ---------------------------------------------------------------------------
__device__ __forceinline__ float block_reduce_wmma(float v, float* smem) {
  const int tid = threadIdx.x;
  smem[tid] = v;
  __syncthreads();
  float r = 0.0f;
  if (tid < 32) {
    v8f c = {};
    v2f ones;
    ones[0] = 1.0f;
    ones[1] = 1.0f;
    const int m = tid & 15;
    const int k0 = (tid >> 4) << 1; // lanes 0-15: K=0,1 ; lanes 16-31: K=2,3
#pragma unroll
    for (int it = 0; it < BDIM / 64; ++it) {
      const float* base = smem + it * 64 + 4 * m + k0;
      v2f a;
      a[0] = base[0];
      a[1] = base[1];
      // (neg_a, A, neg_b, B, c_mod, C, reuse_a, reuse_b)
      c = __builtin_amdgcn_wmma_f32_16x16x4_f32(false, a, false, ones,
                                                (short)0, c, false, false);
    }
    // Lane n (n<16) holds D[0..7][n] in c[0..7]; lane n+16 holds D[8..15][n].
    float t = c[0] + c[1] + c[2] + c[3] + c[4] + c[5] + c[6] + c[7];
    t += __shfl_xor(t, 16, 32); // combine the two halves (wave32)
    r = t;
  }
  return r; // valid for tid < 32 (in particular tid == 0)
}

// ---------------------------------------------------------------------------
// Bilinear corner/weight setup, border clamp, align_corners=True (pixel coords)
// ---------------------------------------------------------------------------
struct Corners {
  int i00, i01, i10, i11;
  float wx, wy;
};

__device__ __forceinline__ Corners make_corners(int x, int y, float fx, float fy) {
  float xs = fminf(fmaxf((float)x + fx, 0.0f), (float)(kW - 1));
  float ys = fminf(fmaxf((float)y + fy, 0.0f), (float)(kH - 1));
  float x0f = floorf(xs);
  float y0f = floorf(ys);
  Corners c;
  c.wx = xs - x0f;
  c.wy = ys - y0f;
  int x0 = (int)x0f;
  int y0 = (int)y0f;
  int x1 = min(x0 + 1, kW - 1);
  int y1 = min(y0 + 1, kH - 1);
  c.i00 = y0 * kW + x0;
  c.i01 = y0 * kW + x1;
  c.i10 = y1 * kW + x0;
  c.i11 = y1 * kW + x1;
  return c;
}

__device__ __forceinline__ float bilerp(const float* __restrict__ p, const Corners& c) {
  float v00 = p[c.i00], v01 = p[c.i01], v10 = p[c.i10], v11 = p[c.i11];
  float top = v00 * (1.0f - c.wx) + v01 * c.wx;
  float bot = v10 * (1.0f - c.wx) + v11 * c.wx;
  return top * (1.0f - c.wy) + bot * c.wy;
}

// ---------------------------------------------------------------------------
// Main kernel: one pixel per loop iteration, 3 channels x 2 images per pixel.
// Flows and gt are single-use streams -> non-temporal loads (keep the 192MB
// L2 for L/R gather reuse). Per-block partial sums go to d_ws (no atomics ->
// bit-deterministic across replays).
// ---------------------------------------------------------------------------
__global__ void __launch_bounds__(BDIM) warp_loss_main(
    const float* __restrict__ L, const float* __restrict__ R,
    const float* __restrict__ flm, const float* __restrict__ frm,
    const float* __restrict__ gt, float* __restrict__ partials, int npix) {
  __shared__ float smem[BDIM];
  float acc = 0.0f;
  const int stride = gridDim.x * BDIM;
  for (int p = blockIdx.x * BDIM + threadIdx.x; p < npix; p += stride) {
    const int b = p / kHW;
    const int r = p - b * kHW;
    const int y = r / kW;
    const int x = r - y * kW;

    const float* fl = flm + (long long)b * 2 * kHW + r;
    const float* fr = frm + (long long)b * 2 * kHW + r;
    const float flx = __builtin_nontemporal_load(fl);
    const float fly = __builtin_nontemporal_load(fl + kHW);
    const float frx = __builtin_nontemporal_load(fr);
    const float fry = __builtin_nontemporal_load(fr + kHW);

    const Corners cl = make_corners(x, y, flx, fly);
    const Corners cr = make_corners(x, y, frx, fry);

    const float* Lb = L + (long long)b * kC * kHW;
    const float* Rb = R + (long long)b * kC * kHW;
    const float* gtp = gt + (long long)b * kC * kHW + r;
#pragma unroll
    for (int c = 0; c < kC; ++c) {
      const float lv = bilerp(Lb + c * kHW, cl);
      const float rv = bilerp(Rb + c * kHW, cr);
      const float g = __builtin_nontemporal_load(gtp + c * kHW);
      acc += fabsf(lv - g) + fabsf(rv - g);
    }
  }
  const float bs = block_reduce_wmma(acc, smem);
  if (threadIdx.x == 0) partials[blockIdx.x] = bs;
}

// ---------------------------------------------------------------------------
// Finalize: one block sums the per-block partials (fixed order -> deterministic)
// through the same WMMA reduction, scales by 1/N and writes the scalar loss.
// ---------------------------------------------------------------------------
__global__ void __launch_bounds__(BDIM) warp_loss_final(
    const float* __restrict__ partials, int n, float* __restrict__ out) {
  __shared__ float smem[BDIM];
  float s = 0.0f;
  for (int i = threadIdx.x; i < n; i += BDIM) s += partials[i];
  const float tot = block_reduce_wmma(s, smem);
  if (threadIdx.x == 0) out[0] = tot * (1.0f / (float)kN);
}

extern "C" void kernel_launch(void* const* d_in, const int* in_sizes, int n_in,
                              void* d_out, int out_size, void* d_ws, size_t ws_size,
                              hipStream_t stream) {
  const float* L   = (const float*)d_in[0];
  const float* R   = (const float*)d_in[1];
  const float* flm = (const float*)d_in[2];
  const float* frm = (const float*)d_in[3];
  const float* gt  = (const float*)d_in[4];
  float* out = (float*)d_out;
  float* partials = (float*)d_ws;

  int nblk = kNBLK; // 7200 blocks * 256 threads * 4 px/thread == 7,372,800 px
  const size_t maxblk = ws_size / sizeof(float);
  if ((size_t)nblk > maxblk) nblk = (int)maxblk; // defensive: fit scratch
  if (nblk < 1) nblk = 1;

  warp_loss_main<<<nblk, BDIM, 0, stream>>>(L, R, flm, frm, gt, partials, kNPIX);
  warp_loss_final<<<1, BDIM, 0, stream>>>(partials, nblk, out);
}